// ColorQuantization_33380485824701
// MI455X (gfx1250) — compile-verified
//
#include <hip/hip_runtime.h>

// CDNA5 / gfx1250, wave32.  V_WMMA_F32_16X16X4_F32: D(16x16) = A(16x4) x B(4x16) + C.
typedef __attribute__((ext_vector_type(2))) float v2f;
typedef __attribute__((ext_vector_type(8))) float v8f;

#define TPW 8   // 16-pixel tiles per wave (amortizes loop-invariant color matrices)

// Nearest-palette-color quantization.
//   score'[c][p] = |c|^2 + 4 - 2*c.p  =  dist^2 - |p|^2 + 4  >= 1  (>0, argmin-equivalent)
// A (M=16 colors of one group, K=4): K = (-2r, -2g, -2b, |c|^2 + 4)   <- loop-invariant
// B (K=4, N=16 pixels):              K = ( r,   g,   b,  1 )
// D[m][n]: lane n holds colors m=0..7 (VGPR r), lane 16+n holds m=8..15, both for pixel n.
// Positive scores => IEEE bits are monotonic as uint => argmin via unsigned min over
// keys (bits & ~63) | color_idx; low idx wins ties (jnp.argmin first-occurrence).
__global__ __launch_bounds__(256)
void ColorQuantization_33380485824701_kernel(const float* __restrict__ patch,   // [3, HW]
                                             const float* __restrict__ colors,  // [64, 3]
                                             float* __restrict__ out,           // [3, HW]
                                             int HW) {
  const int lane = threadIdx.x & 31;
  const int wave = (blockIdx.x * blockDim.x + threadIdx.x) >> 5;
  const int n    = lane & 15;
  const bool hi  = lane >= 16;            // upper half-wave: K2/K3 rows, colors m=8..15
  const int moff = hi ? 8 : 0;            // in-lane color row offset

  // ---- A matrices (colors), loop-invariant. 16x4 f32 A: lanes 0-15={K0,K1}, 16-31={K2,K3}.
  v2f A0, A1, A2, A3;
#pragma unroll
  for (int g = 0; g < 4; ++g) {
    const int c    = g * 16 + n;
    const float cr = colors[3 * c + 0];
    const float cg = colors[3 * c + 1];
    const float cb = colors[3 * c + 2];
    const float cn = cr * cr + cg * cg + cb * cb + 4.0f;   // +4 bias => scores >= 1
    v2f A;
    A.x = hi ? (-2.0f * cb) : (-2.0f * cr);
    A.y = hi ? cn           : (-2.0f * cg);
    if      (g == 0) A0 = A;
    else if (g == 1) A1 = A;
    else if (g == 2) A2 = A;
    else             A3 = A;
  }

  const v8f zero = {};

  for (int t = 0; t < TPW; ++t) {
    const int base = (wave * TPW + t) * 16;
    if (base >= HW) return;
    const int p = base + n;

    // ---- B matrix (pixels): lanes 0-15 = {R,G}, lanes 16-31 = {B,1}. Branchless selects.
    v2f B;
    B.x = patch[p + (hi ? 2 * HW : 0)];
    {
      const float gv = patch[HW + p];
      B.y = hi ? 1.0f : gv;
    }

    // 8 args: (neg_a, A, neg_b, B, c_mod, C, reuse_a, reuse_b)
    v8f D0 = __builtin_amdgcn_wmma_f32_16x16x4_f32(false, A0, false, B, (short)0, zero, false, false);
    v8f D1 = __builtin_amdgcn_wmma_f32_16x16x4_f32(false, A1, false, B, (short)0, zero, false, false);
    v8f D2 = __builtin_amdgcn_wmma_f32_16x16x4_f32(false, A2, false, B, (short)0, zero, false, false);
    v8f D3 = __builtin_amdgcn_wmma_f32_16x16x4_f32(false, A3, false, B, (short)0, zero, false, false);

    // ---- Build 32 sortable keys: (score_bits & ~63) | global_color_idx.
    unsigned k[32];
#pragma unroll
    for (int r = 0; r < 8; ++r) {
      k[r]      = (__float_as_uint(D0[r]) & 0xFFFFFFC0u) | (unsigned)( 0 + moff + r);
      k[8 + r]  = (__float_as_uint(D1[r]) & 0xFFFFFFC0u) | (unsigned)(16 + moff + r);
      k[16 + r] = (__float_as_uint(D2[r]) & 0xFFFFFFC0u) | (unsigned)(32 + moff + r);
      k[24 + r] = (__float_as_uint(D3[r]) & 0xFFFFFFC0u) | (unsigned)(48 + moff + r);
    }
    // ---- 5-level min tree (dep depth 5 instead of 31).
#pragma unroll
    for (int s = 16; s >= 1; s >>= 1) {
#pragma unroll
      for (int j = 0; j < s; ++j) {
        const unsigned a = k[j], b = k[j + s];
        k[j] = (b < a) ? b : a;
      }
    }

    // ---- Single cross-half exchange (one ds_bpermute) + min.
    unsigned kmin = k[0];
    {
      const unsigned ov = (unsigned)__shfl_xor((int)kmin, 16, 32);
      kmin = (ov < kmin) ? ov : kmin;
    }
    const int bi = (int)(kmin & 63u);

    // ---- Coalesced stores: lanes 0-15 write 16 consecutive pixels per channel.
    if (!hi) {
      const int pix = base + n;
      out[pix]          = colors[3 * bi + 0];
      out[HW + pix]     = colors[3 * bi + 1];
      out[2 * HW + pix] = colors[3 * bi + 2];
    }
  }
}

extern "C" void kernel_launch(void* const* d_in, const int* in_sizes, int n_in,
                              void* d_out, int out_size, void* d_ws, size_t ws_size,
                              hipStream_t stream) {
  (void)n_in; (void)out_size; (void)d_ws; (void)ws_size;
  const float* patch  = (const float*)d_in[0];   // [3, 1024, 1024] fp32
  const float* colors = (const float*)d_in[1];   // [64, 3] fp32
  float*       out    = (float*)d_out;           // [1, 3, 1024, 1024] fp32

  const int HW      = in_sizes[0] / 3;           // 1024*1024
  const int waves   = (HW + 16 * TPW - 1) / (16 * TPW);
  const int threads = 256;                       // 8 waves / block
  const int blocks  = (waves * 32 + threads - 1) / threads;

  hipLaunchKernelGGL(ColorQuantization_33380485824701_kernel,
                     dim3(blocks), dim3(threads), 0, stream,
                     patch, colors, out, HW);
}